// GCMCLayer_6322191859868
// MI455X (gfx1250) — compile-verified
//
#include <hip/hip_runtime.h>
#include <hip/hip_bf16.h>

typedef __attribute__((ext_vector_type(2))) float v2f;
typedef __attribute__((ext_vector_type(8))) float v8f;

#define TAU 0.5f
static constexpr int R  = 5;
static constexpr int NU = 50000;
static constexpr int NM = 20000;
static constexpr int E  = 100000;
static constexpr int F  = 4;
static constexpr int D  = 16;
static constexpr int DR = 64;
static constexpr int OUT = 64;
static constexpr long long RE = (long long)R * E;

__device__ __forceinline__ float wred_sum32(float v) {
    v += __shfl_xor(v, 1, 32);
    v += __shfl_xor(v, 2, 32);
    v += __shfl_xor(v, 4, 32);
    v += __shfl_xor(v, 8, 32);
    v += __shfl_xor(v, 16, 32);
    return v;
}
__device__ __forceinline__ float gred_sum8(float v) { // reduce within 8 consecutive lanes
    v += __shfl_xor(v, 1, 32);
    v += __shfl_xor(v, 2, 32);
    v += __shfl_xor(v, 4, 32);
    return v;
}

// ---------------------------------------------------------------------------
// Phase 1: per-edge gate weight w[r,e]; atomic segment-sums into norm_u/norm_m.
// One wave32 per edge; streams review_feat (512MB) with coalesced float4.
// ---------------------------------------------------------------------------
__global__ void edge_weight_kernel(const float* __restrict__ ufk,
                                   const float* __restrict__ mfk,
                                   const float* __restrict__ ufs,
                                   const float* __restrict__ mfs,
                                   const float* __restrict__ rfeat,
                                   const float* __restrict__ proto,
                                   const float* __restrict__ eta,
                                   const int*   __restrict__ eu,
                                   const int*   __restrict__ em,
                                   const int*   __restrict__ kptr,
                                   float* __restrict__ w_out,
                                   float* __restrict__ norm_u,
                                   float* __restrict__ norm_m) {
    const int lane = threadIdx.x & 31;
    const long long widx = (long long)blockIdx.x * 8 + (threadIdx.x >> 5);
    if (widx >= RE) return;
    const int r = (int)(widx / E);
    const long long idx = widx;
    const int u = eu[idx];
    const int m = em[idx];

    // cosine similarity of 16-d node features (full-wave reductions)
    float a = 0.f, b = 0.f;
    if (lane < D) {
        a = ufk[((long long)r * NU + u) * D + lane];
        b = mfk[((long long)r * NM + m) * D + lane];
    }
    const float na = wred_sum32(a * a);
    const float nb = wred_sum32(b * b);
    const float ab = wred_sum32(a * b);
    const float sim_k = ab / (fmaxf(sqrtf(na), 1e-12f) * fmaxf(sqrtf(nb), 1e-12f)) / TAU;

    // sim_all over F=4 groups: 64 floats [f][d], lane loads float2 at lane*2
    const float* us = ufs + ((long long)r * NU + u) * (F * D);
    const float* ms = mfs + ((long long)r * NM + m) * (F * D);
    const float2 u2 = ((const float2*)us)[lane];
    const float2 m2 = ((const float2*)ms)[lane];
    float p = u2.x * m2.x + u2.y * m2.y;       // belongs to f = lane/8
    p = gred_sum8(p);
    float es = __expf(p / TAU);
    es += __shfl_xor(es, 8, 32);
    es += __shfl_xor(es, 16, 32);              // sum over the 4 f-groups
    const float exp_sim = __expf(sim_k) / es;

    // anchor distances: 256 floats [f][dr]; lane loads 8 contiguous floats
    const float* rf = rfeat + ((long long)r * E + (widx % E)) * (F * DR);
    const float4 x0 = ((const float4*)rf)[lane * 2 + 0];
    const float4 x1 = ((const float4*)rf)[lane * 2 + 1];
    const float* pr = proto + (lane >> 3) * DR + (lane & 7) * 8;
    const float4 p0 = ((const float4*)pr)[0];
    const float4 p1 = ((const float4*)pr)[1];
    float q = x0.x * p0.x + x0.y * p0.y + x0.z * p0.z + x0.w * p0.w
            + x1.x * p1.x + x1.y * p1.y + x1.z * p1.z + x1.w * p1.w;
    q = gred_sum8(q);
    const float ad_f = q / TAU;                // all lanes in f-group hold ad_all[f]
    const int k = *kptr;
    const float ad_k = __shfl(ad_f, k * 8, 32);
    float ea = __expf(ad_f);
    ea += __shfl_xor(ea, 8, 32);
    ea += __shfl_xor(ea, 16, 32);
    const float exp_anchor = __expf(ad_k) / ea;

    const float g = 1.f / (1.f + __expf(-eta[idx]));
    const float w = g * exp_anchor + (1.f - g) * exp_sim;
    if (lane == 0) {
        w_out[idx] = w;
        atomicAdd(norm_u + u, w);
        atomicAdd(norm_m + m, w);
    }
}

// ---------------------------------------------------------------------------
// Phase 2: w_norm = w / sqrt(norm_u[u]*norm_m[m]); also emits int_dist output.
// ---------------------------------------------------------------------------
__global__ void wnorm_kernel(const float* __restrict__ w,
                             const float* __restrict__ norm_u,
                             const float* __restrict__ norm_m,
                             const int* __restrict__ eu,
                             const int* __restrict__ em,
                             float* __restrict__ wnorm,
                             float* __restrict__ int_dist) {
    const long long i = (long long)blockIdx.x * blockDim.x + threadIdx.x;
    if (i >= RE) return;
    const float n = sqrtf(norm_u[eu[i]] * norm_m[em[i]]);
    const float v = w[i] / n;
    wnorm[i] = v;
    int_dist[i] = v;
}

// ---------------------------------------------------------------------------
// Phase 3: node projection h = feat @ W^T, (N x 16) @ (16 x 16) per replica.
// One wave per 16-row tile; 4 x V_WMMA_F32_16X16X4_F32 along K=16.
// ---------------------------------------------------------------------------
__global__ void node_proj_kernel(const float* __restrict__ feat,
                                 const float* __restrict__ W,
                                 float* __restrict__ out, int N) {
    const int lane = threadIdx.x & 31;
    const int wv = blockIdx.x * 8 + (threadIdx.x >> 5);
    const int tilesPerR = N / 16;
    if (wv >= R * tilesPerR) return;
    const int r  = wv / tilesPerR;
    const int n0 = (wv % tilesPerR) * 16;
    const int row  = lane & 15;          // A: M index / B: N index
    const int kofs = (lane >> 4) * 2;    // K sub-offset per half-wave
    const float* arow = feat + ((long long)r * N + n0 + row) * D;
    const float* brow = W + ((long long)r * D + row) * D;  // W[r][o=row][d]
    v8f c = {0.f, 0.f, 0.f, 0.f, 0.f, 0.f, 0.f, 0.f};
#pragma unroll
    for (int kk = 0; kk < 4; ++kk) {
        const int k0 = kk * 4 + kofs;
        v2f av; av.x = arow[k0]; av.y = arow[k0 + 1];
        v2f bv; bv.x = brow[k0]; bv.y = brow[k0 + 1];
        c = __builtin_amdgcn_wmma_f32_16x16x4_f32(false, av, false, bv,
                                                  (short)0, c, false, false);
    }
    float* dst = out + ((long long)r * N + n0) * D;
    const int col   = lane & 15;
    const int rbase = (lane & 16) ? 8 : 0;
#pragma unroll
    for (int i = 0; i < 8; ++i)
        dst[(rbase + i) * D + col] = c[i];
}

// ---------------------------------------------------------------------------
// Phase 4: rfp = rf_k @ Wrev/Wfwd^T for 16-edge tiles (16x16x64 -> 16 WMMA
// K-steps, two accumulators share the A operand), then scatter messages:
//   iacc[m] += (hu[u] + rfp_f) * w_norm ;  uacc[u] += (hm[m] + rfp_r) * w_norm
// ---------------------------------------------------------------------------
__global__ void message_kernel(const float* __restrict__ rfeat,
                               const float* __restrict__ wf,
                               const float* __restrict__ wr,
                               const float* __restrict__ hu,
                               const float* __restrict__ hm,
                               const float* __restrict__ wnorm,
                               const int* __restrict__ eu,
                               const int* __restrict__ em,
                               const int* __restrict__ kptr,
                               float* __restrict__ uacc,
                               float* __restrict__ iacc) {
    const int lane = threadIdx.x & 31;
    const int wv = blockIdx.x * 8 + (threadIdx.x >> 5);
    const int tilesPerR = E / 16;
    if (wv >= R * tilesPerR) return;
    const int r  = wv / tilesPerR;
    const int e0 = (wv % tilesPerR) * 16;
    const int k  = *kptr;
    const int row  = lane & 15;
    const int kofs = (lane >> 4) * 2;
    const float* arow = rfeat + (((long long)r * E + e0 + row) * F + k) * DR; // rf_k row
    const float* bfr  = wf + ((long long)r * D + row) * DR;  // W_fwd[o=row][dr]
    const float* brr  = wr + ((long long)r * D + row) * DR;  // W_rev[o=row][dr]
    v8f cf = {0.f, 0.f, 0.f, 0.f, 0.f, 0.f, 0.f, 0.f};
    v8f cr = {0.f, 0.f, 0.f, 0.f, 0.f, 0.f, 0.f, 0.f};
#pragma unroll
    for (int kk = 0; kk < 16; ++kk) {
        const int k0 = kk * 4 + kofs;
        v2f av; av.x = arow[k0]; av.y = arow[k0 + 1];
        v2f b1; b1.x = bfr[k0]; b1.y = bfr[k0 + 1];
        v2f b2; b2.x = brr[k0]; b2.y = brr[k0 + 1];
        cf = __builtin_amdgcn_wmma_f32_16x16x4_f32(false, av, false, b1,
                                                   (short)0, cf, false, false);
        cr = __builtin_amdgcn_wmma_f32_16x16x4_f32(false, av, false, b2,
                                                   (short)0, cr, false, false);
    }
    const int col   = lane & 15;
    const int rbase = (lane & 16) ? 8 : 0;
#pragma unroll
    for (int i = 0; i < 8; ++i) {
        const long long idx = (long long)r * E + e0 + rbase + i;
        const float wn = wnorm[idx];
        const int u = eu[idx];
        const int m = em[idx];
        const float msg_m = (hu[((long long)r * NU + u) * D + col] + cf[i]) * wn;
        const float msg_u = (hm[((long long)r * NM + m) * D + col] + cr[i]) * wn;
        atomicAdd(iacc + (long long)m * D + col, msg_m);
        atomicAdd(uacc + (long long)u * D + col, msg_u);
    }
}

// ---------------------------------------------------------------------------
// Phase 5: out = lrelu(acc) @ fc_w^T + fc_b  ((N x 16) @ (16 x 64))
// ---------------------------------------------------------------------------
__global__ void fc_kernel(const float* __restrict__ acc,
                          const float* __restrict__ fw,
                          const float* __restrict__ fb,
                          float* __restrict__ out, int N) {
    const long long t = (long long)blockIdx.x * blockDim.x + threadIdx.x;
    if (t >= (long long)N * OUT) return;
    const int n = (int)(t >> 6);
    const int o = (int)(t & 63);
    const float4* a4 = (const float4*)(acc + (long long)n * D);
    const float4* w4 = (const float4*)(fw + (long long)o * D);
    float s = fb[o];
#pragma unroll
    for (int j = 0; j < 4; ++j) {
        float4 av = a4[j];
        const float4 wv = w4[j];
        av.x = av.x >= 0.f ? av.x : 0.1f * av.x;
        av.y = av.y >= 0.f ? av.y : 0.1f * av.y;
        av.z = av.z >= 0.f ? av.z : 0.1f * av.z;
        av.w = av.w >= 0.f ? av.w : 0.1f * av.w;
        s += av.x * wv.x + av.y * wv.y + av.z * wv.z + av.w * wv.w;
    }
    out[(long long)n * OUT + o] = s;
}

// ---------------------------------------------------------------------------
extern "C" void kernel_launch(void* const* d_in, const int* in_sizes, int n_in,
                              void* d_out, int out_size, void* d_ws, size_t ws_size,
                              hipStream_t stream) {
    const float* ufk   = (const float*)d_in[0];
    const float* mfk   = (const float*)d_in[1];
    const float* ufs   = (const float*)d_in[2];
    const float* mfs   = (const float*)d_in[3];
    const float* rfeat = (const float*)d_in[4];
    const float* proto = (const float*)d_in[5];
    const float* eta   = (const float*)d_in[6];
    const float* nwf   = (const float*)d_in[7];
    const float* rwf   = (const float*)d_in[8];
    const float* nwr   = (const float*)d_in[9];
    const float* rwr   = (const float*)d_in[10];
    const float* ufcw  = (const float*)d_in[11];
    const float* ufcb  = (const float*)d_in[12];
    const float* ifcw  = (const float*)d_in[13];
    const float* ifcb  = (const float*)d_in[14];
    const int*   eu    = (const int*)d_in[15];
    const int*   em    = (const int*)d_in[16];
    const int*   kptr  = (const int*)d_in[17];

    // workspace layout (bytes)
    char* ws = (char*)d_ws;
    float* w_buf  = (float*)(ws + 0);                         // RE
    float* wn_buf = (float*)(ws + 2000000);                   // RE
    float* norm_u = (float*)(ws + 4000000);                   // NU
    float* norm_m = (float*)(ws + 4200000);                   // NM
    float* hu_buf = (float*)(ws + 4280000);                   // R*NU*16
    float* hm_buf = (float*)(ws + 20280000);                  // R*NM*16
    float* uacc   = (float*)(ws + 26680000);                  // NU*16
    float* iacc   = (float*)(ws + 29880000);                  // NM*16

    float* out_ufeat = (float*)d_out;                         // NU*64
    float* out_ifeat = out_ufeat + (long long)NU * OUT;       // NM*64
    float* out_idist = out_ifeat + (long long)NM * OUT;       // RE

    hipMemsetAsync(norm_u, 0, (size_t)NU * 4, stream);
    hipMemsetAsync(norm_m, 0, (size_t)NM * 4, stream);
    hipMemsetAsync(uacc,   0, (size_t)NU * D * 4, stream);
    hipMemsetAsync(iacc,   0, (size_t)NM * D * 4, stream);

    const int BLK = 256;
    // Phase 1: wave per edge (8 waves/block)
    {
        long long waves = RE;
        int blocks = (int)((waves + 7) / 8);
        edge_weight_kernel<<<blocks, BLK, 0, stream>>>(ufk, mfk, ufs, mfs, rfeat,
            proto, eta, eu, em, kptr, w_buf, norm_u, norm_m);
    }
    // Phase 2
    {
        int blocks = (int)((RE + BLK - 1) / BLK);
        wnorm_kernel<<<blocks, BLK, 0, stream>>>(w_buf, norm_u, norm_m, eu, em,
                                                 wn_buf, out_idist);
    }
    // Phase 3: hu, hm (WMMA f32 16x16x4)
    {
        int tiles = R * (NU / 16);
        node_proj_kernel<<<(tiles + 7) / 8, BLK, 0, stream>>>(ufk, nwf, hu_buf, NU);
        tiles = R * (NM / 16);
        node_proj_kernel<<<(tiles + 7) / 8, BLK, 0, stream>>>(mfk, nwr, hm_buf, NM);
    }
    // Phase 4: messages (WMMA + atomic scatter)
    {
        int tiles = R * (E / 16);
        message_kernel<<<(tiles + 7) / 8, BLK, 0, stream>>>(rfeat, rwf, rwr,
            hu_buf, hm_buf, wn_buf, eu, em, kptr, uacc, iacc);
    }
    // Phase 5: final FCs
    {
        long long tu = (long long)NU * OUT;
        fc_kernel<<<(int)((tu + BLK - 1) / BLK), BLK, 0, stream>>>(uacc, ufcw, ufcb,
                                                                   out_ufeat, NU);
        long long ti = (long long)NM * OUT;
        fc_kernel<<<(int)((ti + BLK - 1) / BLK), BLK, 0, stream>>>(iacc, ifcw, ifcb,
                                                                   out_ifeat, NM);
    }
    (void)in_sizes; (void)n_in; (void)out_size; (void)ws_size;
}